// Dilation_11450382811764
// MI455X (gfx1250) — compile-verified
//
#include <hip/hip_runtime.h>

// 9x9 morphological dilation (sliding-window max), SAME padding, -inf border.
// Input : 16 x 1024 x 1024 x 1 fp32
// Output: same shape.
//
// Memory-bound op (134 MB min HBM traffic ~ 5.8us @ 23.3 TB/s). Strategy:
// single-pass separable max with a 72x72 LDS halo tile per 64x64 output tile
// (read amplification 1.27x => ~152 MB total), halo staged into LDS via the
// CDNA5 async copy path (GLOBAL_LOAD_ASYNC_TO_LDS_B32 + s_wait_asynccnt).
// WMMA is inapplicable: the matrix pipes have no max-plus semiring.

#define IMG_H 1024
#define IMG_W 1024
#define KRAD 4              // radius of 9x9 window
#define TILE 64             // output tile edge
#define HALO (TILE + 2 * KRAD)   // 72

typedef __attribute__((address_space(1))) int g_int;   // global
typedef __attribute__((address_space(3))) int l_int;   // LDS

// ---- CDNA5 async global->LDS copy (one f32 per lane) ------------------------
__device__ __forceinline__ void async_load_f32(const float* g, float* l) {
#if defined(__gfx1250__)
#if __has_builtin(__builtin_amdgcn_global_load_async_to_lds_b32)
  __builtin_amdgcn_global_load_async_to_lds_b32(
      (g_int*)(void*)g, (l_int*)(void*)l, /*offset=*/0, /*cpol=*/0);
#else
  unsigned lds_addr = (unsigned)(unsigned long long)(l_int*)(void*)l;
  asm volatile("global_load_async_to_lds_b32 %0, %1, off"
               :: "v"(lds_addr), "v"(g)
               : "memory");
#endif
#else
  *l = *g;   // host pass / non-gfx1250: plain copy (never codegen'd for device)
#endif
}

__device__ __forceinline__ void wait_async0() {
#if defined(__gfx1250__)
#if __has_builtin(__builtin_amdgcn_s_wait_asynccnt)
  __builtin_amdgcn_s_wait_asynccnt(0);
#else
  asm volatile("s_wait_asynccnt 0" ::: "memory");
#endif
#endif
}

// ---- kernel -----------------------------------------------------------------
// block = (64, 4) = 256 threads = 8 wave32s; grid = (16, 16, batch)
extern "C" __global__ __launch_bounds__(256)
void dilate9x9_kernel(const float* __restrict__ in, float* __restrict__ out) {
  __shared__ float s_in[HALO * HALO];   // 72*72*4 = 20.7 KB  (input + halo)
  __shared__ float s_h [HALO * TILE];   // 72*64*4 = 18.4 KB  (row-max result)

  const int tx  = threadIdx.x;                 // 0..63 (column in tile)
  const int ty  = threadIdx.y;                 // 0..3
  const int tid = ty * TILE + tx;              // 0..255

  const int x0 = blockIdx.x * TILE;
  const int y0 = blockIdx.y * TILE;
  const long long imgBase = (long long)blockIdx.z * (long long)(IMG_H * IMG_W);

  // ---- Stage 1: stage 72x72 halo tile into LDS via async copies ----
  // linear index i: lx = i % 72 (lane-consecutive => coalesced global reads,
  // conflict-free LDS writes), ly = i / 72.
  for (int i = tid; i < HALO * HALO; i += 256) {
    const int lx = i % HALO;
    const int ly = i / HALO;
    const int gx = x0 - KRAD + lx;
    const int gy = y0 - KRAD + ly;
    if ((unsigned)gx < (unsigned)IMG_W && (unsigned)gy < (unsigned)IMG_H) {
      async_load_f32(in + imgBase + (long long)gy * IMG_W + gx, &s_in[i]);
    } else {
      s_in[i] = -__builtin_inff();   // cv2 dilate border value
    }
  }
  wait_async0();        // drain this wave's ASYNCcnt
  __syncthreads();      // make all waves' LDS writes visible

  // ---- Stage 2: horizontal 9-max: s_h[ly][lx] = max(s_in[ly][lx..lx+8]) ----
  // 72 rows x 64 cols; lane-consecutive lx => conflict-free LDS access.
  for (int i = tid; i < HALO * TILE; i += 256) {
    const int lx = i & (TILE - 1);
    const int ly = i >> 6;
    const float* row = &s_in[ly * HALO + lx];
    float m = row[0];
#pragma unroll
    for (int j = 1; j <= 2 * KRAD; ++j) m = fmaxf(m, row[j]);
    s_h[i] = m;         // i == ly*TILE + lx
  }
  __syncthreads();

  // ---- Stage 3: vertical 9-max, log-step reuse, 16 rows per thread ----
  // Thread (tx, ty) owns column tx, output rows [ty*16, ty*16+16).
  const int r0 = ty * 16;

  float h[24];
#pragma unroll
  for (int j = 0; j < 24; ++j) h[j] = s_h[(r0 + j) * TILE + tx];

  float m2[23];
#pragma unroll
  for (int j = 0; j < 23; ++j) m2[j] = fmaxf(h[j], h[j + 1]);
  float m4[21];
#pragma unroll
  for (int j = 0; j < 21; ++j) m4[j] = fmaxf(m2[j], m2[j + 2]);
  float m8[17];
#pragma unroll
  for (int j = 0; j < 17; ++j) m8[j] = fmaxf(m4[j], m4[j + 4]);

  float* o = out + imgBase + (long long)(y0 + r0) * IMG_W + (x0 + tx);
#pragma unroll
  for (int j = 0; j < 16; ++j) {
    // window rows h[j..j+8] = max(m8[j] (covers j..j+7), h[j+8])
    o[(long long)j * IMG_W] = fmaxf(m8[j], h[j + 8]);
  }
}

// ---- launch -----------------------------------------------------------------
extern "C" void kernel_launch(void* const* d_in, const int* in_sizes, int n_in,
                              void* d_out, int out_size, void* d_ws, size_t ws_size,
                              hipStream_t stream) {
  (void)n_in; (void)out_size; (void)d_ws; (void)ws_size;
  const float* in = (const float*)d_in[0];
  float* out      = (float*)d_out;

  const int batch = in_sizes[0] / (IMG_H * IMG_W);   // 16
  dim3 grid(IMG_W / TILE, IMG_H / TILE, batch);      // (16,16,16) = 4096 WGs
  dim3 block(TILE, 4, 1);                            // 256 threads / 8 waves
  dilate9x9_kernel<<<grid, block, 0, stream>>>(in, out);
}